// Motion_Exctractor_Cen_59090160059146
// MI455X (gfx1250) — compile-verified
//
#include <hip/hip_runtime.h>
#include <hip/hip_bf16.h>

// ---------- sizes ----------
#define CN   16      // batch
#define CC   256     // channels
#define CF   32      // frames
#define CWW  784     // 28*28
#define NF   496     // N*(F-1) = 16*31
#define EMB  20
#define C1O  256     // conv1 out channels
#define C2O  512     // conv2 out channels
#define COLS1 4960   // 16*31*10
#define COLS2 2480   // 16*31*5
#define WP1  22      // padded width conv1 input (20+2)
#define WP2  12      // padded width conv2 input (10+2)
#define BNEPS 1e-5f
#define AROW 772     // 768 + 4 pad: lane stride mod 64 banks = 4 -> conflict-free

typedef float v2f __attribute__((ext_vector_type(2)));
typedef float v8f __attribute__((ext_vector_type(8)));
typedef int   v4i __attribute__((vector_size(16)));

#if defined(__has_builtin)
#if __has_builtin(__builtin_amdgcn_global_load_async_to_lds_b128)
#define HAVE_ASYNC_LDS 1
#endif
#endif

#ifdef HAVE_ASYNC_LDS
#define AS1 __attribute__((address_space(1)))
#define AS3 __attribute__((address_space(3)))
#endif

// ---------------------------------------------------------------
// Stage A: one streaming pass over h (411 MB). Per (n,c,f) compute
// s1=Σx, si=Σx*i, sj=Σx*j, s2=Σx². One block per (n,c,f).
// ---------------------------------------------------------------
__global__ __launch_bounds__(256) void k_stageA(const float* __restrict__ h,
                                                float4* __restrict__ stats) {
  const int b = blockIdx.x;            // (n*CC+c)*CF+f
  const int tid = threadIdx.x;
  const float* src = h + (size_t)b * CWW;
  float s1 = 0.f, si = 0.f, sj = 0.f, s2 = 0.f;
  if (tid < 196) {                     // 196 float4 = 784 floats
    float4 v = ((const float4*)src)[tid];
    float vv[4] = {v.x, v.y, v.z, v.w};
#pragma unroll
    for (int q = 0; q < 4; ++q) {
      int pix = tid * 4 + q;
      int i = pix / 28;
      int j = pix - i * 28;
      float x = vv[q];
      s1 += x; si += x * (float)i; sj += x * (float)j; s2 += x * x;
    }
  }
  __shared__ float red[4][256];
  red[0][tid] = s1; red[1][tid] = si; red[2][tid] = sj; red[3][tid] = s2;
  __syncthreads();
  for (int off = 128; off > 0; off >>= 1) {
    if (tid < off) {
      red[0][tid] += red[0][tid + off];
      red[1][tid] += red[1][tid + off];
      red[2][tid] += red[2][tid + off];
      red[3][tid] += red[3][tid + off];
    }
    __syncthreads();
  }
  if (tid == 0) stats[b] = make_float4(red[0][0], red[1][0], red[2][0], red[3][0]);
}

// ---------------------------------------------------------------
// Stage B: per-channel BN0 scale. Mean/bias cancel in the frame
// diff, so only g*rsqrt(var+eps) is needed. One wave32 per channel.
// ---------------------------------------------------------------
__global__ __launch_bounds__(256) void k_stageB(const float4* __restrict__ stats,
                                                const float* __restrict__ g0,
                                                float* __restrict__ scale) {
  const int lane = threadIdx.x & 31;
  const int c = blockIdx.x * 8 + (threadIdx.x >> 5);
  float S1 = 0.f, S2 = 0.f;
  for (int e = lane; e < CN * CF; e += 32) {   // 512 (n,f) entries
    int n = e >> 5, f = e & 31;
    float4 st = stats[(n * CC + c) * CF + f];
    S1 += st.x; S2 += st.w;
  }
#pragma unroll
  for (int off = 16; off > 0; off >>= 1) {
    S1 += __shfl_down(S1, off, 32);
    S2 += __shfl_down(S2, off, 32);
  }
  if (lane == 0) {
    const float inv = 1.0f / (float)(CN * CF * CWW);   // 401408
    float mean = S1 * inv;
    float var  = S2 * inv - mean * mean;
    scale[c] = g0[c] * rsqrtf(var + BNEPS);
  }
}

// ---------------------------------------------------------------
// Build padded, TRANSPOSED conv1 input mp_t[nf][22][ci]  (ci fastest):
// diff = scale_ci*(rawcen[f+1]-rawcen[f]); m = diff @ lin_w.T + lin_b
// borders (w=0,21) zeroed for the stride-2 (1,3) conv. Writes coalesced.
// ---------------------------------------------------------------
__global__ __launch_bounds__(256) void k_mp(const float4* __restrict__ stats,
                                            const float* __restrict__ scale,
                                            const float* __restrict__ lin_w,
                                            const float* __restrict__ lin_b,
                                            float* __restrict__ mp) {
  int idx = blockIdx.x * 256 + threadIdx.x;    // ((nf*WP1 + w)*256 + ci)
  int ci = idx & 255;
  int t  = idx >> 8;
  int w  = t % WP1;
  int nf = t / WP1;
  float val = 0.f;
  if (w >= 1 && w <= EMB) {
    int e = w - 1;
    int n = nf / 31, f = nf % 31;
    float4 s0 = stats[(n * CC + ci) * CF + f];
    float4 s1 = stats[(n * CC + ci) * CF + f + 1];
    float sc = scale[ci];
    float di = sc * (s1.y - s0.y);
    float dj = sc * (s1.z - s0.z);
    val = di * lin_w[e * 2 + 0] + dj * lin_w[e * 2 + 1] + lin_b[e];
  }
  mp[idx] = val;
}

// ---------------------------------------------------------------
// Conv as 3 accumulated GEMMs on V_WMMA_F32_16X16X4_F32.
// Block = 8 waves; LDS stages A = weights[16co][768] (row-padded to 772).
// Staged with GLOBAL_LOAD_ASYNC_TO_LDS_B128 (ASYNCcnt) when available.
// Each wave owns a strip of 4 col tiles (16co x 64col): one A fragment
// feeds 4 WMMAs; B comes from the ci-fastest activation layout as a
// single global_load_b64 per WMMA.
// ---------------------------------------------------------------
template <int WPAD, int WO, int NCOLS, int NTILES>
__global__ __launch_bounds__(256) void k_conv_wmma(const float* __restrict__ Wt,
                                                   const float* __restrict__ Bpt,
                                                   float* __restrict__ Z) {
  __shared__ float A[16][AROW];                // ~48 KB
  const int tid = threadIdx.x;
  const int co0 = blockIdx.y * 16;

  // Stage A tile: A[row][i] = Wt[(co0+row)*768 + i], i = ci*3+kx
#ifdef HAVE_ASYNC_LDS
  for (int chunk = tid; chunk < 3072; chunk += 256) {   // 3072 x 16B
    int row = chunk / 192, c4 = chunk - row * 192;
    float* g = const_cast<float*>(Wt + (size_t)(co0 + row) * 768 + c4 * 4);
    float* l = &A[row][c4 * 4];
    __builtin_amdgcn_global_load_async_to_lds_b128((AS1 v4i*)g, (AS3 v4i*)l, 0, 0);
  }
  asm volatile("s_wait_asynccnt 0" ::: "memory");
#else
  for (int chunk = tid; chunk < 3072; chunk += 256) {
    int row = chunk / 192, c4 = chunk - row * 192;
    const float4 v = *(const float4*)(Wt + (size_t)(co0 + row) * 768 + c4 * 4);
    *(float4*)&A[row][c4 * 4] = v;
  }
#endif
  __syncthreads();

  const int wv = tid >> 5, lane = tid & 31;
  const int m16  = lane & 15;                  // co row / col-in-tile
  const int half = lane >> 4;                  // k-half selector
  const int t0 = __builtin_amdgcn_readfirstlane(blockIdx.x * 8 + wv) * 4;
  if (t0 >= NTILES) return;                    // wave-uniform exit
  int nt = NTILES - t0; nt = nt > 4 ? 4 : nt;  // scalar tile count (1..4)

  // per-tile B base pointers: Bpt[(nf*WPAD + 2*wo + kx)*256 + ci]
  const float* bb[4];
  int colv[4];
#pragma unroll
  for (int t = 0; t < 4; ++t) {
    int col = (t0 + t) * 16 + m16;
    int nf = col / WO, wo = col - nf * WO;
    colv[t] = col;
    bb[t] = Bpt + ((size_t)nf * WPAD + 2 * wo) * 256;
  }

  v8f acc[4];
#pragma unroll
  for (int t = 0; t < 4; ++t) acc[t] = (v8f){0.f,0.f,0.f,0.f,0.f,0.f,0.f,0.f};

#pragma unroll
  for (int kx = 0; kx < 3; ++kx) {
    const float* albase = &A[m16][kx];         // index with ci*3
    for (int cib = 0; cib < 256; cib += 4) {
      const int ci0 = cib + 2 * half;
      v2f a;
      a.x = albase[ci0 * 3];                   // W[co0+m16][ci0][kx]
      a.y = albase[ci0 * 3 + 3];               // W[co0+m16][ci0+1][kx]
#pragma unroll
      for (int t = 0; t < 4; ++t) {
        if (t < nt) {                          // scalar branch (nt uniform)
          v2f b = *(const v2f*)(bb[t] + (size_t)kx * 256 + ci0);
          acc[t] = __builtin_amdgcn_wmma_f32_16x16x4_f32(
              false, a, false, b, (short)0, acc[t], false, false);
        }
      }
    }
  }

  // C/D layout: VGPR r -> M = r + 8*half; N = lane%16
  const int mrow = co0 + 8 * half;
#pragma unroll
  for (int t = 0; t < 4; ++t) {
    if (t < nt) {
#pragma unroll
      for (int r = 0; r < 8; ++r)
        Z[(size_t)(mrow + r) * NCOLS + colv[t]] = acc[t][r];
    }
  }
}

// ---------------------------------------------------------------
// BN stats over the col dimension per out-channel -> (scale, shift)
// ---------------------------------------------------------------
__global__ __launch_bounds__(256) void k_bnstat(const float* __restrict__ Z,
                                                const float* __restrict__ g,
                                                const float* __restrict__ bta,
                                                float2* __restrict__ ss, int ncols) {
  const int c = blockIdx.x, tid = threadIdx.x;
  const float* row = Z + (size_t)c * ncols;
  float s = 0.f, s2 = 0.f;
  for (int i = tid; i < ncols; i += 256) { float x = row[i]; s += x; s2 += x * x; }
  __shared__ float r1[256], r2[256];
  r1[tid] = s; r2[tid] = s2;
  __syncthreads();
  for (int off = 128; off > 0; off >>= 1) {
    if (tid < off) { r1[tid] += r1[tid + off]; r2[tid] += r2[tid + off]; }
    __syncthreads();
  }
  if (tid == 0) {
    float inv = 1.0f / (float)ncols;
    float mean = r1[0] * inv;
    float var  = r2[0] * inv - mean * mean;
    float sc = g[c] * rsqrtf(var + BNEPS);
    ss[c] = make_float2(sc, bta[c] - mean * sc);
  }
}

// tanh(bn1(z1)) into padded TRANSPOSED conv2 input y1p[nf][12][ci]
__global__ __launch_bounds__(256) void k_y1p(const float* __restrict__ z1,
                                             const float2* __restrict__ ss,
                                             float* __restrict__ y1p) {
  int idx = blockIdx.x * 256 + threadIdx.x;    // ((nf*WP2 + w)*256 + ci)
  int ci = idx & 255;
  int t  = idx >> 8;
  int w  = t % WP2;
  int nf = t / WP2;
  float v = 0.f;
  if (w >= 1 && w <= 10) {
    float2 s = ss[ci];
    v = tanhf(z1[(size_t)ci * COLS1 + nf * 10 + (w - 1)] * s.x + s.y);
  }
  y1p[idx] = v;
}

// tanh(bn2(z2)) + AvgPool(1,5) -> out [N, 512, 31, 1]
__global__ __launch_bounds__(256) void k_final(const float* __restrict__ z2,
                                               const float2* __restrict__ ss,
                                               float* __restrict__ out) {
  int idx = blockIdx.x * 256 + threadIdx.x;    // ((n*512+co)*31+f)
  int f  = idx % 31;
  int t  = idx / 31;
  int co = t % C2O;
  int n  = t / C2O;
  float2 s = ss[co];
  const float* row = z2 + (size_t)co * COLS2 + (n * 31 + f) * 5;
  float acc = 0.f;
#pragma unroll
  for (int w = 0; w < 5; ++w) acc += tanhf(row[w] * s.x + s.y);
  out[idx] = acc * 0.2f;
}

// ---------------------------------------------------------------
extern "C" void kernel_launch(void* const* d_in, const int* in_sizes, int n_in,
                              void* d_out, int out_size, void* d_ws, size_t ws_size,
                              hipStream_t stream) {
  const float* h       = (const float*)d_in[0];
  const float* bn0_g   = (const float*)d_in[1];
  // d_in[2] bn0_b: cancels in the frame diff
  const float* lin_w   = (const float*)d_in[3];
  const float* lin_b   = (const float*)d_in[4];
  const float* conv1_w = (const float*)d_in[5];
  const float* bn1_g   = (const float*)d_in[6];
  const float* bn1_b   = (const float*)d_in[7];
  const float* conv2_w = (const float*)d_in[8];
  const float* bn2_g   = (const float*)d_in[9];
  const float* bn2_b   = (const float*)d_in[10];

  float* ws = (float*)d_ws;
  float4* stats  = (float4*)ws;                         size_t off = 4 * (size_t)CN * CC * CF; // 524288
  float*  scale  = ws + off;                            off += 256;
  float*  mp     = ws + off;                            off += (size_t)CC * NF * WP1;          // 2793472
  float*  z1     = ws + off;                            off += (size_t)C1O * COLS1;            // 1269760
  float2* ss1    = (float2*)(ws + off);                 off += 512;
  float*  y1p    = ws + off;                            off += (size_t)CC * NF * WP2;          // 1523712
  float*  z2     = ws + off;                            off += (size_t)C2O * COLS2;            // 1269760
  float2* ss2    = (float2*)(ws + off);

  // 1) streaming reduction over h (bandwidth-bound: ~411 MB)
  k_stageA<<<CN * CC * CF, 256, 0, stream>>>(h, stats);
  // 2) BN0 per-channel scale
  k_stageB<<<CC / 8, 256, 0, stream>>>(stats, bn0_g, scale);
  // 3) fused diff + Linear(2->EMB), padded transposed layout
  k_mp<<<(CC * NF * WP1) / 256, 256, 0, stream>>>(stats, scale, lin_w, lin_b, mp);
  // 4) conv1 as WMMA GEMM: [256 x 768] * [768 x 4960]; 310 tiles -> 78 strips
  k_conv_wmma<WP1, 10, COLS1, 310><<<dim3(10, 16), 256, 0, stream>>>(conv1_w, mp, z1);
  // 5) BN1 stats
  k_bnstat<<<C1O, 256, 0, stream>>>(z1, bn1_g, bn1_b, ss1, COLS1);
  // 6) tanh + pad (transposed)
  k_y1p<<<(CC * NF * WP2) / 256, 256, 0, stream>>>(z1, ss1, y1p);
  // 7) conv2 as WMMA GEMM: [512 x 768] * [768 x 2480]; 155 tiles -> 39 strips
  k_conv_wmma<WP2, 5, COLS2, 155><<<dim3(5, 32), 256, 0, stream>>>(conv2_w, y1p, z2);
  // 8) BN2 stats
  k_bnstat<<<C2O, 256, 0, stream>>>(z2, bn2_g, bn2_b, ss2, COLS2);
  // 9) tanh + avgpool -> out
  k_final<<<(CN * C2O * 31) / 256, 256, 0, stream>>>(z2, ss2, (float*)d_out);
}